// MotionAwareStateSpaceModel_17841294147831
// MI455X (gfx1250) — compile-verified
//
#include <hip/hip_runtime.h>
#include <hip/hip_bf16.h>
#include <stdint.h>

#define B_SZ     2
#define LM       8192
#define BT_SZ    8
#define LS       16384
#define D_MODEL  256
#define D_STATE  16
#define D_INNER  512
#define DT_RANK  16
#define M_ROWS   (B_SZ * LM)     /* 16384 */
#define S_ROWS   (BT_SZ * LS)    /* 131072 */
#define LDS_PAD  4

typedef float v2f __attribute__((ext_vector_type(2)));
typedef float v8f __attribute__((ext_vector_type(8)));

__device__ __forceinline__ float sigmoidf_(float x) { return 1.f / (1.f + __expf(-x)); }
__device__ __forceinline__ float siluf_(float x)    { return x * sigmoidf_(x); }

// Async copy of 16 bytes global -> LDS (ASYNCcnt-tracked, CDNA5).
// lds_off = raw LDS byte offset (low 32 bits of generic pointer), g = global src.
__device__ __forceinline__ void async_g2lds_b128(unsigned lds_off, const float* g) {
    asm volatile("global_load_async_to_lds_b128 %0, %1, off"
                 :: "v"(lds_off), "v"(g) : "memory");
}
__device__ __forceinline__ void wait_asynccnt0() {
    asm volatile("s_wait_asynccnt 0x0" ::: "memory");
}

// ---------------------------------------------------------------------------
// LayerNorm over last dim (256) ; one wave32 per row, 8 elements per lane
// ---------------------------------------------------------------------------
__global__ void ln_rows_kernel(const float* __restrict__ x,
                               const float* __restrict__ w,
                               const float* __restrict__ b,
                               float* __restrict__ out, int nrows) {
    int wavesPerBlock = blockDim.x >> 5;
    int row  = blockIdx.x * wavesPerBlock + (threadIdx.x >> 5);
    int lane = threadIdx.x & 31;
    if (row >= nrows) return;
    const float* xr = x + (long)row * D_MODEL;
    float v[8];
    float s = 0.f;
#pragma unroll
    for (int i = 0; i < 8; ++i) { v[i] = xr[lane + i * 32]; s += v[i]; }
#pragma unroll
    for (int off = 16; off >= 1; off >>= 1) s += __shfl_xor(s, off, 32);
    float mu = s * (1.f / D_MODEL);
    float var = 0.f;
#pragma unroll
    for (int i = 0; i < 8; ++i) { float d = v[i] - mu; var += d * d; }
#pragma unroll
    for (int off = 16; off >= 1; off >>= 1) var += __shfl_xor(var, off, 32);
    var *= (1.f / D_MODEL);
    float rstd = rsqrtf(var + 1e-5f);
    float* orow = out + (long)row * D_MODEL;
#pragma unroll
    for (int i = 0; i < 8; ++i) {
        int c = lane + i * 32;
        orow[c] = (v[i] - mu) * rstd * w[c] + b[c];
    }
}

// ---------------------------------------------------------------------------
// Staged fp32 WMMA GEMM:  C[M,N] = A[M,K] * W^T   (W is [N,K] row-major)
// Requires lda == K (contiguous 16xK A panel per C tile-row).
//   - Block = 8 waves; grid.x = M/16; block covers 8*TN*16 columns at
//     column-block blockIdx.y.
//   - A panel (16xK) staged into LDS via GLOBAL_LOAD_ASYNC_TO_LDS_B128,
//     completed with s_wait_asynccnt + barrier; rows padded by 4 floats so
//     the 16-lane ds_load_b64 fragment reads are bank-conflict-free.
//   - Each wave computes TN 16x16 tiles reusing one A fragment per k-step.
// ---------------------------------------------------------------------------
template <int TN, int K>
__global__ void gemm_f32_wmma_staged(const float* __restrict__ A,
                                     const float* __restrict__ W,
                                     float* __restrict__ C, int ldc, int N) {
    __shared__ float As[16 * (K + LDS_PAD)];
    const int tid  = threadIdx.x;
    const int wv   = tid >> 5;
    const int lane = tid & 31;
    const int tm   = blockIdx.x;

    // ---- async stage of the contiguous 16*K A panel ----
    const float* Ap = A + (long)tm * 16 * K;
    constexpr int VPR  = K / 4;            // float4 vectors per row
    constexpr int NVEC = 16 * VPR;
    for (int i = tid; i < NVEC; i += 256) {
        int r = i / VPR;
        int c4 = i - r * VPR;
        unsigned lds_off =
            (unsigned)(uintptr_t)(&As[r * (K + LDS_PAD) + c4 * 4]);
        async_g2lds_b128(lds_off, Ap + i * 4);
    }
    wait_asynccnt0();
    __syncthreads();

    // ---- compute ----
    const int half = lane >> 4;            // K-pair select within 16x16x4 frag
    const int l15  = lane & 15;
    const int tnb  = (blockIdx.y * 8 + wv) * TN;   // first 16-col tile index

    const float* wrow[TN];
    v8f acc[TN];
#pragma unroll
    for (int j = 0; j < TN; ++j) {
        wrow[j] = W + (long)((tnb + j) * 16 + l15) * K;
        acc[j] = v8f{};
        __builtin_prefetch(wrow[j], 0, 3);
    }
    const float* arow = &As[l15 * (K + LDS_PAD)];
#pragma unroll 2
    for (int k = 0; k < K; k += 4) {
        int ka = k + half * 2;
        v2f a;
        a.x = arow[ka];
        a.y = arow[ka + 1];
#pragma unroll
        for (int j = 0; j < TN; ++j) {
            v2f bb;
            bb.x = wrow[j][ka];
            bb.y = wrow[j][ka + 1];
            acc[j] = __builtin_amdgcn_wmma_f32_16x16x4_f32(
                false, a, false, bb, (short)0, acc[j], false, false);
        }
    }
#pragma unroll
    for (int j = 0; j < TN; ++j) {
        int ccol = (tnb + j) * 16 + l15;
#pragma unroll
        for (int i = 0; i < 8; ++i) {
            int crow = tm * 16 + i + half * 8;
            C[(long)crow * ldc + ccol] = acc[j][i];
        }
    }
}

// ---------------------------------------------------------------------------
// Simple fp32 WMMA GEMM (one wave per 16x16 tile) for small/odd shapes.
// ---------------------------------------------------------------------------
__global__ void gemm_f32_wmma_kernel(const float* __restrict__ A, int lda,
                                     const float* __restrict__ W, int ldw,
                                     float* __restrict__ C, int ldc,
                                     int M, int N, int K) {
    int wave = (int)((blockIdx.x * blockDim.x + threadIdx.x) >> 5);
    int lane = threadIdx.x & 31;
    int tilesN = N >> 4;
    int tm = wave / tilesN;
    int tn = wave - tm * tilesN;
    if (tm * 16 >= M) return;                 // whole-wave uniform exit
    int half = lane >> 4;
    int l15  = lane & 15;
    const float* arow = A + (long)(tm * 16 + l15) * lda;
    const float* wrow = W + (long)(tn * 16 + l15) * ldw;
    v8f acc = {};
    for (int k = 0; k < K; k += 4) {
        int ka = k + half * 2;
        v2f a, bb;
        a.x  = arow[ka];     a.y  = arow[ka + 1];
        bb.x = wrow[ka];     bb.y = wrow[ka + 1];
        acc = __builtin_amdgcn_wmma_f32_16x16x4_f32(
            false, a, false, bb, (short)0, acc, false, false);
    }
    int ccol = tn * 16 + l15;
#pragma unroll
    for (int j = 0; j < 8; ++j) {
        int crow = tm * 16 + j + half * 8;
        C[(long)crow * ldc + ccol] = acc[j];
    }
}

// ---------------------------------------------------------------------------
// Fused: multi depthwise conv (K=3, pad 1) + sigmoid gate, single depthwise
// conv computed ON THE FLY at gathered rows + silu, combine, final silu.
//   mxz layout: [m, 1024]  (cols 0..511 = mx, 512..1023 = z)
//   sxp layout: [s, 512]
// ---------------------------------------------------------------------------
__global__ void fused_conv_gate_kernel(const float* __restrict__ mxz,
                                       const float* __restrict__ sxp,
                                       const int*   __restrict__ tn_inverse,
                                       const float* __restrict__ conv_w,
                                       const float* __restrict__ conv_b,
                                       const float* __restrict__ conv_s_w,
                                       const float* __restrict__ conv_s_b,
                                       float* __restrict__ mga) {
    int m = blockIdx.x;
    int d = threadIdx.x;
    int l = m & (LM - 1);
    float w0 = conv_w[d * 3 + 0], w1 = conv_w[d * 3 + 1], w2 = conv_w[d * 3 + 2];
    float xm1 = (l > 0)      ? mxz[(long)(m - 1) * 1024 + d] : 0.f;
    float x0  =                mxz[(long)m * 1024 + d];
    float xp1 = (l < LM - 1) ? mxz[(long)(m + 1) * 1024 + d] : 0.f;
    float mco = w0 * xm1 + w1 * x0 + w2 * xp1 + conv_b[d];
    int src = tn_inverse[m];
    int ls  = src & (LS - 1);
    float s0w = conv_s_w[d * 3 + 0], s1w = conv_s_w[d * 3 + 1], s2w = conv_s_w[d * 3 + 2];
    float sm1 = (ls > 0)      ? sxp[(long)(src - 1) * D_INNER + d] : 0.f;
    float sc0 =                 sxp[(long)src * D_INNER + d];
    float sp1 = (ls < LS - 1) ? sxp[(long)(src + 1) * D_INNER + d] : 0.f;
    float sco = s0w * sm1 + s1w * sc0 + s2w * sp1 + conv_s_b[d];
    float sf  = siluf_(sco);
    float g   = sigmoidf_(mco) * sf + mco;
    mga[(long)m * D_INNER + d] = siluf_(g);
}

// ---------------------------------------------------------------------------
// Selective scan: 16 lanes per (b,d) channel (one per state n), 2 channels
// per wave32. Sequential over l. y_out = (scan_y + u*D) * silu(z).
//   xdbl: [m, 48]  (0..15 = dt-rank part, 16..31 = B, 32..47 = C)
// ---------------------------------------------------------------------------
__global__ void selective_scan_kernel(const float* __restrict__ mga,
                                      const float* __restrict__ delta,
                                      const float* __restrict__ xdbl,
                                      const float* __restrict__ mxz,
                                      const float* __restrict__ A_log,
                                      const float* __restrict__ D_param,
                                      const float* __restrict__ dt_bias,
                                      float* __restrict__ ys) {
    int wave = (int)((blockIdx.x * blockDim.x + threadIdx.x) >> 5);
    int lane = threadIdx.x & 31;
    int half = lane >> 4;
    int n    = lane & 15;
    int ch   = wave * 2 + half;
    if (ch >= B_SZ * D_INNER) return;
    int b = ch >> 9;
    int d = ch & (D_INNER - 1);
    float a  = -__expf(A_log[d * D_STATE + n]);
    float Dp = D_param[d];
    float db = dt_bias[d];
    float st = 0.f;
    long base = (long)b * LM;
    for (int l = 0; l < LM; ++l) {
        long m = base + l;
        float dtv = delta[m * D_INNER + d] + db;
        dtv = (dtv > 20.f) ? dtv : log1pf(__expf(dtv));      // softplus
        float u  = mga[m * D_INNER + d];
        float Bn = xdbl[m * 48 + DT_RANK + n];
        float Cn = xdbl[m * 48 + DT_RANK + D_STATE + n];
        st = __expf(dtv * a) * st + dtv * u * Bn;
        float y = st * Cn;
        y += __shfl_xor(y, 1, 32);
        y += __shfl_xor(y, 2, 32);
        y += __shfl_xor(y, 4, 32);
        y += __shfl_xor(y, 8, 32);                           // stays in half
        if (n == 0) {
            float z = mxz[m * 1024 + D_INNER + d];
            ys[m * D_INNER + d] = (y + u * Dp) * siluf_(z);
        }
    }
}

// ---------------------------------------------------------------------------
static inline int gemm_blocks(int M, int N) {
    long waves = (long)(M / 16) * (N / 16);
    long threads = waves * 32;
    return (int)((threads + 255) / 256);
}

extern "C" void kernel_launch(void* const* d_in, const int* in_sizes, int n_in,
                              void* d_out, int out_size, void* d_ws, size_t ws_size,
                              hipStream_t stream) {
    const float* multi_x    = (const float*)d_in[0];
    const float* single_x   = (const float*)d_in[1];
    const int*   tn_inverse = (const int*)  d_in[2];
    // d_in[3]=single_idx, d_in[4]=multi_idx : identity permutations, unused
    const float* ln_w       = (const float*)d_in[5];
    const float* ln_b       = (const float*)d_in[6];
    const float* in_proj_w  = (const float*)d_in[7];   // [1024, 256]
    const float* conv_w     = (const float*)d_in[8];   // [512,1,3]
    const float* conv_b     = (const float*)d_in[9];
    const float* conv_s_w   = (const float*)d_in[10];
    const float* conv_s_b   = (const float*)d_in[11];
    const float* x_proj_w   = (const float*)d_in[12];  // [48, 512]
    const float* dt_w       = (const float*)d_in[13];  // [512, 16]
    const float* dt_b       = (const float*)d_in[14];
    const float* out_proj_w = (const float*)d_in[15];  // [256, 512]
    const float* A_log      = (const float*)d_in[16];  // [512, 16]
    const float* D_param    = (const float*)d_in[17];
    float* out = (float*)d_out;

    float* ws  = (float*)d_ws;
    // live buffers
    float* mh   = ws;                       // 16384*256          = 4,194,304
    float* sh   = ws + 4194304;             // 131072*256         -> end 37,748,736
    float* mxz  = ws + 37748736;            // 16384*1024         -> end 54,525,952
    float* sxp  = ws + 54525952;            // 131072*512         -> end 121,634,816
    // aliases into the (dead after GEMMs) mh/sh region
    float* mga   = ws;                      // 16384*512 = 8,388,608
    float* xdbl  = ws + 8388608;            // 16384*48  -> 9,175,040
    float* delta = ws + 9175040;            // 16384*512 -> 17,563,648
    float* ysb   = ws + 17563648;           // 16384*512 -> 25,952,256

    // 1) LayerNorm both paths
    ln_rows_kernel<<<M_ROWS / 8, 256, 0, stream>>>(multi_x, ln_w, ln_b, mh, M_ROWS);
    ln_rows_kernel<<<S_ROWS / 8, 256, 0, stream>>>(single_x, ln_w, ln_b, sh, S_ROWS);

    // 2) in_proj GEMMs (async-LDS-staged fp32 WMMA)
    gemm_f32_wmma_staged<4, D_MODEL><<<dim3(M_ROWS / 16, 2), 256, 0, stream>>>(
        mh, in_proj_w, mxz, 2 * D_INNER, 2 * D_INNER);
    gemm_f32_wmma_staged<4, D_MODEL><<<dim3(S_ROWS / 16, 1), 256, 0, stream>>>(
        sh, in_proj_w, sxp, D_INNER, D_INNER);

    // 3) fused convs + gather + gating -> mga
    fused_conv_gate_kernel<<<M_ROWS, D_INNER, 0, stream>>>(
        mxz, sxp, tn_inverse, conv_w, conv_b, conv_s_w, conv_s_b, mga);

    // 4) x_proj:  xdbl[16384,48] = mga @ x_proj_w^T  (small-N path)
    gemm_f32_wmma_kernel<<<gemm_blocks(M_ROWS, DT_RANK + 2 * D_STATE), 256, 0, stream>>>(
        mga, D_INNER, x_proj_w, D_INNER, xdbl, DT_RANK + 2 * D_STATE,
        M_ROWS, DT_RANK + 2 * D_STATE, D_INNER);

    // 5) dt projection: delta[16384,512] = xdbl[:, :16] @ dt_w^T   (K=16)
    gemm_f32_wmma_kernel<<<gemm_blocks(M_ROWS, D_INNER), 256, 0, stream>>>(
        xdbl, DT_RANK + 2 * D_STATE, dt_w, DT_RANK, delta, D_INNER,
        M_ROWS, D_INNER, DT_RANK);

    // 6) selective scan -> ys (includes *silu(z) and +u*D)
    {
        int waves = (B_SZ * D_INNER + 1) / 2;          // 512 waves
        int threads = waves * 32;
        selective_scan_kernel<<<(threads + 255) / 256, 256, 0, stream>>>(
            mga, delta, xdbl, mxz, A_log, D_param, dt_b, ysb);
    }

    // 7) out_proj (staged, K=512): out[16384,256] = ys @ out_proj_w^T
    gemm_f32_wmma_staged<2, D_INNER><<<dim3(M_ROWS / 16, 1), 256, 0, stream>>>(
        ysb, out_proj_w, out, D_MODEL, D_MODEL);
}